// RankingModule_4123168604522
// MI455X (gfx1250) — compile-verified
//
#include <hip/hip_runtime.h>

typedef float v2f __attribute__((ext_vector_type(2)));
typedef float v8f __attribute__((ext_vector_type(8)));

#define SDIM 4096
#define DDIM 128
#define HDIM 12
#define BDIM 2
#define NBINS 10

#define ROWS_TOTAL (HDIM * SDIM)          // 49152 rows per batch to reduce
#define ROWGROUPS 64
#define ROWS_PER_WAVE (ROWS_TOTAL / ROWGROUPS)  // 768
#define STEPS (ROWS_PER_WAVE / 4)         // 192 WMMAs per wave
#define JGROUPS (SDIM / 16)               // 256
#define WAVES_PER_BLOCK 8
#define TOTAL_WAVES (BDIM * JGROUPS * ROWGROUPS) // 32768

// ---------------------------------------------------------------- kernel 0
__global__ void zero_ws_k(float* ws, int n) {
    for (int i = blockIdx.x * blockDim.x + threadIdx.x; i < n;
         i += gridDim.x * blockDim.x)
        ws[i] = 0.0f;
}

// ---------------------------------------------------------------- kernel 1
// Column sums via V_WMMA_F32_16X16X4_F32 with A = ones(16x4):
// D = A*B + C accumulates per-column sums of a 4x16 tile of scores.
// B layout (32-bit, 4x16): VGPR0 = {row k=0 lanes 0-15 | row k=2 lanes 16-31},
//                          VGPR1 = {row k=1            | row k=3           }.
__global__ void colsum_wmma_k(const float* __restrict__ scores,
                              float* __restrict__ s_ws) {
    const int wave = blockIdx.x * WAVES_PER_BLOCK + (threadIdx.x >> 5);
    const int lane = threadIdx.x & 31;
    const int b    = wave >> 14;           // / (JGROUPS*ROWGROUPS)
    const int rem  = wave & 16383;
    const int jg   = rem >> 6;             // column group (16 cols)
    const int rg   = rem & 63;             // row group (768 rows)

    const size_t rowStride = SDIM;
    const float* p = scores
        + (size_t)b * HDIM * SDIM * SDIM
        + ((size_t)(rg * ROWS_PER_WAVE) + (size_t)((lane >> 4) << 1)) * rowStride
        + (size_t)(jg * 16 + (lane & 15));

    v2f a; a[0] = 1.0f; a[1] = 1.0f;
    v8f c0 = {}; v8f c1 = {};

    #pragma unroll 4
    for (int t = 0; t < STEPS; t += 2) {
        const float* q0 = p + (size_t)(4 * t) * rowStride;
        const float* q1 = q0 + 4 * rowStride;
        v2f b0, b1;
        b0[0] = __builtin_nontemporal_load(q0);
        b0[1] = __builtin_nontemporal_load(q0 + rowStride);
        b1[0] = __builtin_nontemporal_load(q1);
        b1[1] = __builtin_nontemporal_load(q1 + rowStride);
        c0 = __builtin_amdgcn_wmma_f32_16x16x4_f32(false, a, false, b0,
                                                   (short)0, c0, false, false);
        c1 = __builtin_amdgcn_wmma_f32_16x16x4_f32(false, a, false, b1,
                                                   (short)0, c1, false, false);
    }
    v8f cs = c0 + c1;
    // All 16 rows of D hold identical column sums (A is all-ones); row 0,
    // lanes 0-15 cover N = 0..15.
    if (lane < 16)
        atomicAdd(&s_ws[b * SDIM + jg * 16 + lane], cs[0]);
}

// ---------------------------------------------------------------- kernel 2
__global__ void minmax_k(const float* __restrict__ s_ws, float* mm) {
    __shared__ float smn[256], smx[256];
    const int t = threadIdx.x;
    float mn = 3.402823466e38f, mx = -3.402823466e38f;
    for (int i = t; i < BDIM * SDIM; i += 256) {
        float v = s_ws[i];
        mn = fminf(mn, v);
        mx = fmaxf(mx, v);
    }
    smn[t] = mn; smx[t] = mx;
    __syncthreads();
    for (int d = 128; d > 0; d >>= 1) {
        if (t < d) {
            smn[t] = fminf(smn[t], smn[t + d]);
            smx[t] = fmaxf(smx[t], smx[t + d]);
        }
        __syncthreads();
    }
    if (t == 0) { mm[0] = smn[0]; mm[1] = smx[0]; }
}

// ---------------------------------------------------------------- kernel 3
// Digitize batch-0 scores into bins 1..10, stable-compact indices per bin
// (pad -1) directly into the int32 tail of d_out. One block, 1024 threads,
// 4 elements per thread, Hillis-Steele scan over per-thread bin counts.
__global__ void __launch_bounds__(1024)
bins_scan_k(const float* __restrict__ s_ws, int* __restrict__ idx_out) {
    __shared__ int lcnt[1024][NBINS];
    const int t = threadIdx.x;

    const float smin = s_ws[BDIM * SDIM];
    const float smax = s_ws[BDIM * SDIM + 1];
    const float inv  = 1.0f / (smax - smin);

    float bnd[NBINS + 1];
    #pragma unroll
    for (int k = 0; k <= NBINS; ++k)
        bnd[k] = (float)((double)k * (1.01 / 10.0)); // np.linspace(0,1.01,11)

    // -1 padding for the whole indices output
    for (int i = t; i < NBINS * SDIM; i += 1024) idx_out[i] = -1;

    #pragma unroll
    for (int b = 0; b < NBINS; ++b) lcnt[t][b] = 0;

    int mybin[4];
    #pragma unroll
    for (int e = 0; e < 4; ++e) {
        const int j = t * 4 + e;
        const float x = (s_ws[j] - smin) * inv;  // batch 0 only
        int bin = 0;                              // digitize = #boundaries <= x
        #pragma unroll
        for (int k = 0; k <= NBINS; ++k) bin += (x >= bnd[k]) ? 1 : 0;
        bin = bin < 1 ? 1 : (bin > NBINS ? NBINS : bin);
        mybin[e] = bin - 1;
        lcnt[t][bin - 1]++;
    }
    __syncthreads();

    // inclusive scan over threads for each bin
    for (int d = 1; d < 1024; d <<= 1) {
        int v[NBINS];
        if (t >= d) {
            #pragma unroll
            for (int b = 0; b < NBINS; ++b) v[b] = lcnt[t - d][b];
        }
        __syncthreads();
        if (t >= d) {
            #pragma unroll
            for (int b = 0; b < NBINS; ++b) lcnt[t][b] += v[b];
        }
        __syncthreads();
    }

    #pragma unroll
    for (int e = 0; e < 4; ++e) {
        const int bin  = mybin[e];
        const int base = (t == 0) ? 0 : lcnt[t - 1][bin]; // exclusive prefix
        int loc = 0;
        #pragma unroll
        for (int e2 = 0; e2 < 4; ++e2)
            if (e2 < e && mybin[e2] == bin) ++loc;
        idx_out[bin * SDIM + base + loc] = t * 4 + e;
    }
}

// ---------------------------------------------------------------- kernel 4
// patches[bin][b][pos][d] = idx>=0 ? patch[b][idx][d] : 0
__global__ void gather_k(const float* __restrict__ patch,
                         const int* __restrict__ idx_out,
                         float* __restrict__ out) {
    const int rowId = blockIdx.x * 8 + (threadIdx.x >> 5);
    const int lane  = threadIdx.x & 31;
    const int bin   = rowId >> 13;
    const int b     = (rowId >> 12) & 1;
    const int pos   = rowId & (SDIM - 1);
    const int idx   = idx_out[bin * SDIM + pos];

    float4 v = make_float4(0.f, 0.f, 0.f, 0.f);
    if (idx >= 0)
        v = *(const float4*)(patch + ((size_t)(b * SDIM + idx) * DDIM) + lane * 4);
    *(float4*)(out + (size_t)rowId * DDIM + lane * 4) = v;
}

// ---------------------------------------------------------------- launcher
extern "C" void kernel_launch(void* const* d_in, const int* in_sizes, int n_in,
                              void* d_out, int out_size, void* d_ws, size_t ws_size,
                              hipStream_t stream) {
    const float* scores = (const float*)d_in[0];           // [2,12,4096,4096]
    const float* patch  = (const float*)d_in[1];           // [2,4096,128]
    float* out = (float*)d_out;
    float* ws  = (float*)d_ws;                             // 8192 partials + 2 mm
    int* idx_out = (int*)(out + (size_t)NBINS * BDIM * SDIM * DDIM);

    zero_ws_k<<<32, 256, 0, stream>>>(ws, BDIM * SDIM + 2);
    colsum_wmma_k<<<TOTAL_WAVES / WAVES_PER_BLOCK, 256, 0, stream>>>(scores, ws);
    minmax_k<<<1, 256, 0, stream>>>(ws, ws + BDIM * SDIM);
    bins_scan_k<<<1, 1024, 0, stream>>>(ws, idx_out);
    gather_k<<<NBINS * BDIM * SDIM / 8, 256, 0, stream>>>(patch, idx_out, out);
}